// pRNN_76562087018544
// MI455X (gfx1250) — compile-verified
//
#include <hip/hip_runtime.h>

typedef __attribute__((ext_vector_type(2))) float v2f;
typedef __attribute__((ext_vector_type(8))) float v8f;

#define N_LAYERS 96
#define BATCH    8192
#define DD       64
#define BT       128        // batch rows per block (8 waves x 16 rows)
#define G_STRIDE 68         // padded LDS row stride for gathered tile (bank-conflict-free A frags)
#define W_STRIDE 65         // padded LDS row stride for W^T

__global__ __launch_bounds__(256) void prnn_last_layer_kernel(
    const float* __restrict__ x,
    const float* __restrict__ w_in,
    const float* __restrict__ b_in,
    const float* __restrict__ W,
    const float* __restrict__ Bi,
    const float* __restrict__ h,
    const int*   __restrict__ src_layer,
    const int*   __restrict__ src_feat,
    float*       __restrict__ out)
{
    __shared__ float        g_lds[BT * G_STRIDE];
    __shared__ float        wt_lds[DD * W_STRIDE];   // wt[k][o] = W[95][o][k]
    __shared__ float        bi_lds[DD];
    __shared__ const float* ptr_lds[DD];             // per-d gather base (element at b=0)
    __shared__ float        scale_lds[DD];
    __shared__ float        bias_lds[DD];
    __shared__ int          flag_lds[DD];            // 1 => apply relu (layer-0 source)

    const int tid = threadIdx.x;
    const int b0  = blockIdx.x * BT;

    // ---- Phase 0a: gather descriptors for the 64 K-columns of layer 95 ----
    if (tid < DD) {
        const int d  = tid;
        const int sl = src_layer[(N_LAYERS - 1) * DD + d];
        const int sf = src_feat [(N_LAYERS - 1) * DD + d];
        if (sl == 0) {
            // g[b,d] = relu(x[b,sf] * w_in[sf] + b_in[sf])
            ptr_lds[d]   = x + sf;
            scale_lds[d] = w_in[sf];
            bias_lds[d]  = b_in[sf];
            flag_lds[d]  = 1;
        } else {
            // g[b,d] = h[sl, b, sf]
            ptr_lds[d]   = h + (size_t)sl * BATCH * DD + sf;
            scale_lds[d] = 1.0f;
            bias_lds[d]  = 0.0f;
            flag_lds[d]  = 0;
        }
        bi_lds[d] = Bi[(N_LAYERS - 1) * DD + d];
    }

    // ---- Phase 0b: stage W[95] transposed into LDS (coalesced global read) ----
    {
        const float* Wl = W + (size_t)(N_LAYERS - 1) * DD * DD;
        for (int i = tid; i < DD * DD; i += 256) {
            const int o = i / DD;            // output feature
            const int k = i % DD;            // reduction dim
            wt_lds[k * W_STRIDE + o] = Wl[i];
        }
    }
    __syncthreads();

    // ---- Phase 1: gather g tile (BT x 64) with fused affine/ReLU ----
    // linear = d*BT + r : consecutive lanes take consecutive batch rows of the
    // same (layer,feat) stream -> one stride-256B stream per half-wave.
    for (int it = 0; it < (BT * DD) / 256; ++it) {
        const int linear = tid + 256 * it;
        const int d = linear / BT;
        const int r = linear % BT;
        const float* p = ptr_lds[d];
        float val = p[(size_t)(b0 + r) * DD];
        val = fmaf(val, scale_lds[d], bias_lds[d]);
        if (flag_lds[d]) val = fmaxf(val, 0.0f);
        g_lds[r * G_STRIDE + d] = val;
    }
    __syncthreads();

    // ---- Phase 2: WMMA fp32 GEMM. Wave w -> batch rows [w*16, w*16+16), all 64 outs ----
    const int wave  = tid >> 5;              // wave32
    const int lane  = tid & 31;
    const int m     = lane & 15;             // row (A) / col (B,D) within tile
    const int khi   = (lane >> 4) << 1;      // 0 or 2: K sub-offset per A/B layout
    const int rbase = wave * 16;

    v8f acc0 = {}, acc1 = {}, acc2 = {}, acc3 = {};

    #pragma unroll
    for (int K0 = 0; K0 < DD; K0 += 4) {
        // A fragment: lane holds g[rbase+m][K0+khi+{0,1}]
        const float* ga = &g_lds[(rbase + m) * G_STRIDE + K0 + khi];
        v2f a; a.x = ga[0]; a.y = ga[1];

        // B fragments: lane holds wt[K0+khi+{0,1}][n0+m]
        const float* wb = &wt_lds[(K0 + khi) * W_STRIDE];
        v2f bf;

        bf.x = wb[ 0 + m]; bf.y = wb[W_STRIDE +  0 + m];
        acc0 = __builtin_amdgcn_wmma_f32_16x16x4_f32(false, a, false, bf, (short)0, acc0, false, false);
        bf.x = wb[16 + m]; bf.y = wb[W_STRIDE + 16 + m];
        acc1 = __builtin_amdgcn_wmma_f32_16x16x4_f32(false, a, false, bf, (short)0, acc1, false, false);
        bf.x = wb[32 + m]; bf.y = wb[W_STRIDE + 32 + m];
        acc2 = __builtin_amdgcn_wmma_f32_16x16x4_f32(false, a, false, bf, (short)0, acc2, false, false);
        bf.x = wb[48 + m]; bf.y = wb[W_STRIDE + 48 + m];
        acc3 = __builtin_amdgcn_wmma_f32_16x16x4_f32(false, a, false, bf, (short)0, acc3, false, false);
    }

    // ---- Epilogue: bias + ReLU + store. D layout: lane,vgpr v -> M=v+8*(lane>=16), N=n0+m ----
    const int mrow = (lane >> 4) << 3;       // 0 or 8
    float* outp = out + (size_t)(b0 + rbase) * DD;
    const float bv0 = bi_lds[ 0 + m];
    const float bv1 = bi_lds[16 + m];
    const float bv2 = bi_lds[32 + m];
    const float bv3 = bi_lds[48 + m];

    #pragma unroll
    for (int v = 0; v < 8; ++v) {
        const int row = mrow + v;
        outp[row * DD +  0 + m] = fmaxf(acc0[v] + bv0, 0.0f);
        outp[row * DD + 16 + m] = fmaxf(acc1[v] + bv1, 0.0f);
        outp[row * DD + 32 + m] = fmaxf(acc2[v] + bv2, 0.0f);
        outp[row * DD + 48 + m] = fmaxf(acc3[v] + bv3, 0.0f);
    }
}

extern "C" void kernel_launch(void* const* d_in, const int* in_sizes, int n_in,
                              void* d_out, int out_size, void* d_ws, size_t ws_size,
                              hipStream_t stream) {
    (void)in_sizes; (void)n_in; (void)out_size; (void)d_ws; (void)ws_size;
    const float* x         = (const float*)d_in[0];
    const float* w_in      = (const float*)d_in[1];
    const float* b_in      = (const float*)d_in[2];
    const float* W         = (const float*)d_in[3];
    const float* Bi        = (const float*)d_in[4];
    const float* h         = (const float*)d_in[5];
    const int*   src_layer = (const int*)d_in[6];
    const int*   src_feat  = (const int*)d_in[7];
    float*       out       = (float*)d_out;

    dim3 grid(BATCH / BT);   // 64 blocks
    dim3 block(256);         // 8 wave32
    prnn_last_layer_kernel<<<grid, block, 0, stream>>>(
        x, w_in, b_in, W, Bi, h, src_layer, src_feat, out);
}